// FSOL_29892972380538
// MI455X (gfx1250) — compile-verified
//
#include <hip/hip_runtime.h>
#include <math.h>

#define THETA 0.7f

typedef __attribute__((ext_vector_type(2))) float v2f;
typedef __attribute__((ext_vector_type(8))) float v8f;

// Problem constants (from reference setup_inputs)
// B=2, C=256, H=W=64 (HW=4096), N=9 taps, Odc=256, Ocdc=256, out=[B,512,64,64]

// ---------------- transpose x [B,256,4096] -> xt [B,4096,256] ----------------
__global__ void FSOL_transpose(const float* __restrict__ x, float* __restrict__ xt) {
  __shared__ float tile[32][33];
  const int b   = blockIdx.z;
  const int hw0 = blockIdx.x * 32;
  const int c0  = blockIdx.y * 32;
  const float* xb  = x  + (size_t)b * 256 * 4096;
  float*       xtb = xt + (size_t)b * 4096 * 256;
  for (int j = threadIdx.y; j < 32; j += 8)
    tile[j][threadIdx.x] = xb[(size_t)(c0 + j) * 4096 + hw0 + threadIdx.x];
  __syncthreads();
  for (int j = threadIdx.y; j < 32; j += 8)
    xtb[(size_t)(hw0 + j) * 256 + c0 + threadIdx.x] = tile[threadIdx.x][j];
}

// ------------- weight prep: deform  Wdc[(n*256+c)*256+o] = dconv_w[o,c,n] -------------
__global__ void FSOL_wprep_dc(const float* __restrict__ dw, float* __restrict__ W) {
  int i = blockIdx.x * blockDim.x + threadIdx.x;        // 2304*256
  if (i >= 2304 * 256) return;
  int o = i & 255, k = i >> 8;
  int n = k >> 8, c = k & 255;                           // k = n*256 + c
  W[i] = dw[(size_t)o * 2304 + c * 9 + n];
}

// ------------- weight prep: CDC, center tap fuses -THETA*sum --------------------------
// tap order t: 0=up(x[h-1,w]) 1=left 2=center 3=right 4=down ; raw cdc_w[o, c, 0, t]
__global__ void FSOL_wprep_cdc(const float* __restrict__ cw, float* __restrict__ W) {
  int i = blockIdx.x * blockDim.x + threadIdx.x;        // 1280*256
  if (i >= 1280 * 256) return;
  int o = i & 255, k = i >> 8;
  int t = k >> 8, c = k & 255;                           // k = t*256 + c
  const float* base = cw + (size_t)o * 1280 + c * 5;
  float v = base[t];
  if (t == 2) v -= THETA * (base[0] + base[1] + base[2] + base[3] + base[4]);
  W[i] = v;
}

// ------------- weight prep: offset(18) + modulation(9) conv, padded to 32 ocs ---------
__global__ void FSOL_wprep_om(const float* __restrict__ pw, const float* __restrict__ mw,
                              float* __restrict__ W) {
  int i = blockIdx.x * blockDim.x + threadIdx.x;        // 2304*32
  if (i >= 2304 * 32) return;
  int o = i & 31, k = i >> 5;
  int g = k >> 8, c = k & 255;                           // k = g*256 + c, g = kh*3+kw
  float v = 0.f;
  if (o < 18)      v = pw[(size_t)o * 2304 + c * 9 + g];
  else if (o < 27) v = mw[(size_t)(o - 18) * 2304 + c * 9 + g];
  W[i] = v;
}

__global__ void FSOL_bias(const float* __restrict__ pb, const float* __restrict__ mb,
                          float* __restrict__ bias) {
  int o = threadIdx.x;
  if (o < 32) bias[o] = (o < 18) ? pb[o] : ((o < 27) ? mb[o - 18] : 0.f);
}

// ------------- gather table: 3x3 im2col (G=9,T=1) -------------------------------------
__global__ void FSOL_tab_om(int* __restrict__ tI, float* __restrict__ tW) {
  int i = blockIdx.x * blockDim.x + threadIdx.x;        // 2*4096*9
  if (i >= 2 * 4096 * 9) return;
  int g = i % 9, hw = (i / 9) & 4095;
  int h = hw >> 6, w = hw & 63;
  int hh = h + g / 3 - 1, ww = w + g % 3 - 1;
  bool in = (hh >= 0) & (hh < 64) & (ww >= 0) & (ww < 64);
  tI[i] = in ? hh * 64 + ww : 0;
  tW[i] = in ? 1.f : 0.f;
}

// ------------- gather table: CDC plus-stencil (G=5,T=1) -------------------------------
__global__ void FSOL_tab_cdc(int* __restrict__ tI, float* __restrict__ tW) {
  int i = blockIdx.x * blockDim.x + threadIdx.x;        // 2*4096*5
  if (i >= 2 * 4096 * 5) return;
  int t = i % 5, hw = (i / 5) & 4095;
  int h = hw >> 6, w = hw & 63;
  const int dh[5] = {-1, 0, 0, 0, 1};
  const int dw[5] = { 0,-1, 0, 1, 0};
  int hh = h + dh[t], ww = w + dw[t];
  bool in = (hh >= 0) & (hh < 64) & (ww >= 0) & (ww < 64);
  tI[i] = in ? hh * 64 + ww : 0;
  tW[i] = in ? 1.f : 0.f;
}

// ------------- gather table: deformable bilinear corners (G=9,T=4), mod folded in -----
__global__ void FSOL_tab_dc(const float* __restrict__ om,   // [b][27][4096]
                            int* __restrict__ tI, float* __restrict__ tW) {
  int i = blockIdx.x * blockDim.x + threadIdx.x;        // 2*4096*9
  if (i >= 2 * 4096 * 9) return;
  int n = i % 9, hw = (i / 9) & 4095, b = i / (9 * 4096);
  int h = hw >> 6, w = hw & 63;
  const float* omb = om + (size_t)b * 27 * 4096;
  float offx = omb[(size_t)n * 4096 + hw];
  float offy = omb[(size_t)(9 + n) * 4096 + hw];
  float mval = omb[(size_t)(18 + n) * 4096 + hw];      // already sigmoided
  float px = offx + (float)(n / 3 - 1) + (float)(h + 1);
  float py = offy + (float)(n % 3 - 1) + (float)(w + 1);
  float fx = floorf(px), fy = floorf(py);
  float qx0 = fminf(fmaxf(fx,        0.f), 65.f);
  float qy0 = fminf(fmaxf(fy,        0.f), 65.f);
  float qx1 = fminf(fmaxf(fx + 1.f,  0.f), 65.f);
  float qy1 = fminf(fmaxf(fy + 1.f,  0.f), 65.f);
  float pxc = fminf(fmaxf(px, 0.f), 65.f);
  float pyc = fminf(fmaxf(py, 0.f), 65.f);
  float g_lt = (1.f + (qx0 - pxc)) * (1.f + (qy0 - pyc));
  float g_rb = (1.f - (qx1 - pxc)) * (1.f - (qy1 - pyc));
  float g_lb = (1.f + (qx0 - pxc)) * (1.f - (qy1 - pyc));
  float g_rt = (1.f - (qx1 - pxc)) * (1.f + (qy0 - pyc));
  float qx[4] = {qx0, qx1, qx0, qx1};
  float qy[4] = {qy0, qy1, qy1, qy0};
  float gg[4] = {g_lt, g_rb, g_lb, g_rt};
  int base = i * 4;
  #pragma unroll
  for (int j = 0; j < 4; ++j) {
    int ix = (int)qx[j], iy = (int)qy[j];
    bool in = (ix >= 1) & (ix <= 64) & (iy >= 1) & (iy <= 64);  // padded border == 0
    tI[base + j] = in ? (ix - 1) * 64 + (iy - 1) : 0;
    tW[base + j] = in ? gg[j] * mval : 0.f;
  }
}

// ---------------- fused unfold + WMMA f32 GEMM, MT-way M register tiling --------------
// out[b, o, hw] = sum_{g,c} ( sum_j tW[m,g,j] * xt[b, tI[m,g,j], c] ) * Wt[g*256+c, o]
// EPI: 0 = relu -> d_out ch[0,256)   1 = relu -> d_out ch[256,512)
//      2 = +bias, sigmoid for oc>=18 -> offmod[b][27][4096]
// Each wave reuses one B fragment for MT WMMAs (MT accumulators, MT A tiles in LDS).
template <int G, int T, int OC, int EPI, int MT>
__global__ __launch_bounds__((OC / 16) * 32)
void FSOL_gemm(const float* __restrict__ xt, const float* __restrict__ Wt,
               const int* __restrict__ tabI, const float* __restrict__ tabW,
               const float* __restrict__ bias, float* __restrict__ out) {
  static_assert(EPI == 2 || OC == 256, "fast epilogue assumes OC==256");
  static_assert(EPI != 2 || MT == 1, "sigmoid epilogue assumes MT==1");
  constexpr int NT    = (OC / 16) * 32;  // threads per block
  constexpr int ROWS  = MT * 16;         // spatial rows per block
  constexpr int TILES = 4096 / ROWS;     // hw tiles per batch image
  constexpr int TPR   = NT / ROWS;       // threads per A row in the build phase
  constexpr int CPT   = 256 / TPR;       // channels per thread in the build phase
  __shared__ float As[ROWS][260];        // K-chunk of A, padded (conflict-free frag reads)

  const int b    = blockIdx.x / TILES;
  const int hw0  = (blockIdx.x % TILES) * ROWS;
  const int tid  = threadIdx.x;
  const int lane = tid & 31;
  const int wave = tid >> 5;
  const int col  = lane & 15;            // N (and A's M) index within fragment
  const int kup  = (lane >> 4) << 1;     // K sub-row: 0 (lanes 0-15) / 2 (lanes 16-31)
  const int ocb  = wave * 16;
  const int am   = tid / TPR;            // A row this thread builds
  const int ac0  = (tid % TPR) * CPT;

  const float* xb = xt + (size_t)b * 4096 * 256;
  v8f acc[MT];
  #pragma unroll
  for (int mt = 0; mt < MT; ++mt) acc[mt] = {};

  for (int g = 0; g < G; ++g) {
    // ---- build A[ROWS][256] for this K-group: weighted sum of T gathered rows of xt
    int ii[T]; float wwv[T];
    const int tb = ((((b << 12) + hw0 + am) * G) + g) * T;
    #pragma unroll
    for (int j = 0; j < T; ++j) { ii[j] = tabI[tb + j]; wwv[j] = tabW[tb + j]; }
    for (int c = ac0; c < ac0 + CPT; c += 4) {
      float ax = 0.f, ay = 0.f, az = 0.f, aw = 0.f;
      #pragma unroll
      for (int j = 0; j < T; ++j) {
        const float4 v = *(const float4*)(xb + (size_t)ii[j] * 256 + c);
        ax += wwv[j] * v.x; ay += wwv[j] * v.y; az += wwv[j] * v.z; aw += wwv[j] * v.w;
      }
      As[am][c + 0] = ax; As[am][c + 1] = ay; As[am][c + 2] = az; As[am][c + 3] = aw;
    }
    __syncthreads();
    // ---- 64 WMMA K-steps over this group; B fragment shared across MT m-tiles
    const float* Wg = Wt + ((size_t)g * 256 + kup) * OC + ocb + col;
    #pragma unroll 4
    for (int k0 = 0; k0 < 256; k0 += 4) {
      v2f bv = { Wg[(size_t)k0 * OC], Wg[(size_t)(k0 + 1) * OC] };   // B: lane=N, K rows
      #pragma unroll
      for (int mt = 0; mt < MT; ++mt) {
        v2f av = { As[mt * 16 + col][k0 + kup], As[mt * 16 + col][k0 + kup + 1] };
        acc[mt] = __builtin_amdgcn_wmma_f32_16x16x4_f32(
            false, av, false, bv, (short)0, acc[mt], false, false);
      }
    }
    __syncthreads();
  }

  const int rbase = (lane >> 4) << 3;    // C/D: lanes 16-31 hold M=r+8
  if constexpr (EPI == 2) {
    const int oc = ocb + col;
    if (oc < 27) {
      const float bv = bias[oc];
      #pragma unroll
      for (int r = 0; r < 8; ++r) {
        float v = acc[0][r] + bv;
        if (oc >= 18) v = 1.f / (1.f + expf(-v));    // sigmoid for modulation channels
        out[((size_t)(b * 27 + oc)) * 4096 + hw0 + rbase + r] = v;
      }
    }
  } else {
    // relu; per m-tile: transpose through LDS so each thread stores 32B contiguous
    float* Ls = &As[0][0];                           // 4096 floats per pass
    const int choff = (EPI == 1) ? 256 : 0;
    #pragma unroll
    for (int mt = 0; mt < MT; ++mt) {
      #pragma unroll
      for (int r = 0; r < 8; ++r)
        Ls[(ocb + col) * 16 + rbase + r] = fmaxf(acc[mt][r], 0.f);
      __syncthreads();
      const int oc = tid >> 1;
      const int m  = (tid & 1) * 8;
      const float4* Lv = (const float4*)Ls;
      float* dst = out + ((size_t)(b * 512 + choff + oc)) * 4096 + hw0 + mt * 16 + m;
      *(float4*)(dst + 0) = Lv[tid * 2 + 0];
      *(float4*)(dst + 4) = Lv[tid * 2 + 1];
      __syncthreads();
    }
  }
}

extern "C" void kernel_launch(void* const* d_in, const int* in_sizes, int n_in,
                              void* d_out, int out_size, void* d_ws, size_t ws_size,
                              hipStream_t stream) {
  const float* x     = (const float*)d_in[0];
  const float* p_w   = (const float*)d_in[1];
  const float* p_b   = (const float*)d_in[2];
  const float* m_w   = (const float*)d_in[3];
  const float* m_b   = (const float*)d_in[4];
  const float* dc_w  = (const float*)d_in[5];
  const float* cdc_w = (const float*)d_in[6];
  float* out = (float*)d_out;

  char* wsp = (char*)d_ws;
  size_t off = 0;
  auto alloc = [&](size_t bytes) -> void* {
    void* p = wsp + off;
    off = (off + bytes + 255) & ~(size_t)255;
    return p;
  };
  float* xt      = (float*)alloc((size_t)2 * 4096 * 256 * 4);   // 8.4 MB
  float* Wdc     = (float*)alloc((size_t)2304 * 256 * 4);       // 2.36 MB
  float* Wcdc    = (float*)alloc((size_t)1280 * 256 * 4);       // 1.31 MB
  float* Wom     = (float*)alloc((size_t)2304 * 32 * 4);
  float* bias    = (float*)alloc(32 * 4);
  float* offmod  = (float*)alloc((size_t)2 * 27 * 4096 * 4);
  int*   tOmI    = (int*)  alloc((size_t)2 * 4096 * 9 * 4);
  float* tOmW    = (float*)alloc((size_t)2 * 4096 * 9 * 4);
  int*   tDcI    = (int*)  alloc((size_t)2 * 4096 * 9 * 4 * 4);
  float* tDcW    = (float*)alloc((size_t)2 * 4096 * 9 * 4 * 4);
  int*   tCdI    = (int*)  alloc((size_t)2 * 4096 * 5 * 4);
  float* tCdW    = (float*)alloc((size_t)2 * 4096 * 5 * 4);

  // 1) channel-last transpose of x (coalesced gather rows for the GEMMs)
  FSOL_transpose<<<dim3(128, 8, 2), dim3(32, 8), 0, stream>>>(x, xt);
  // 2) weight / bias / static-table prep (independent)
  FSOL_wprep_dc <<<2304, 256, 0, stream>>>(dc_w, Wdc);
  FSOL_wprep_cdc<<<1280, 256, 0, stream>>>(cdc_w, Wcdc);
  FSOL_wprep_om <<<288, 256, 0, stream>>>(p_w, m_w, Wom);
  FSOL_bias     <<<1, 32, 0, stream>>>(p_b, m_b, bias);
  FSOL_tab_om   <<<288, 256, 0, stream>>>(tOmI, tOmW);
  FSOL_tab_cdc  <<<160, 256, 0, stream>>>(tCdI, tCdW);
  // 3) offset + modulation conv as WMMA GEMM (27 ocs padded to 32), sigmoid epilogue
  FSOL_gemm<9, 1, 32, 2, 1><<<512, 64, 0, stream>>>(xt, Wom, tOmI, tOmW, bias, offmod);
  // 4) bilinear corner tables (modulation folded into weights)
  FSOL_tab_dc<<<288, 256, 0, stream>>>(offmod, tDcI, tDcW);
  // 5) deformable conv GEMM: M=8192, N=256, K=2304, 2-way M tiling -> ch [0,256)
  FSOL_gemm<9, 4, 256, 0, 2><<<256, 512, 0, stream>>>(xt, Wdc, tDcI, tDcW, nullptr, out);
  // 6) CDC conv GEMM: M=8192, N=256, K=1280 (diff term pre-fused)   -> ch [256,512)
  FSOL_gemm<5, 1, 256, 1, 2><<<256, 512, 0, stream>>>(xt, Wcdc, tCdI, tCdW, nullptr, out);
}